// SkipGram_89464168776162
// MI455X (gfx1250) — compile-verified
//
#include <hip/hip_runtime.h>
#include <stdint.h>

// ---------------------------------------------------------------------------
// SkipGram forward: segment-summed input embeddings + output/negative gathers.
// Pure bandwidth problem (0.06 FLOP/byte) -> pure data-movement optimization:
//   * out_vec/neg_vec: async DMA copies global -> LDS -> global (CDNA5
//     GLOBAL_LOAD_ASYNC_TO_LDS_B128 / GLOBAL_STORE_ASYNC_FROM_LDS_B128,
//     ASYNCcnt-tracked, no VGPR round trip), fully unrolled issue bursts,
//     non-temporal stores (outputs are write-once; keep 192MB L2 for tables).
//   * in_vec: wave-per-segment binary search over sorted segment_ids,
//     float4 (b128) accumulate (deterministic order), global_prefetch_b8,
//     non-temporal result stores.
// ---------------------------------------------------------------------------

typedef __attribute__((ext_vector_type(4))) float v4f;   // native clang vector

namespace {
constexpr int    kEmb        = 300;
constexpr int    kBatch      = 16384;
constexpr int    kTotal      = 65536;
constexpr int    kNg         = 5;
constexpr int    kChunks     = kEmb / 4;            // 75 float4 per row
constexpr int    kThirdLanes = kChunks - 64;        // lanes 0..10 own chunk 3
constexpr int    kRowBytes   = kEmb * 4;            // 1200 B, 16B aligned
constexpr int    kOutRows    = kBatch;              // 16384
constexpr int    kNegRows    = kBatch * kNg;        // 81920
constexpr int    kGatherRows = kOutRows + kNegRows; // 98304
constexpr size_t kOutBase    = (size_t)kBatch * kEmb;      // 4,915,200 floats
constexpr size_t kNegBase    = 2 * (size_t)kBatch * kEmb;  // 9,830,400 floats

constexpr int kWavesPerBlock = 8;   // 256 threads, wave32
constexpr int kRowsPerWave   = 4;   // rows copied per async batch
constexpr int kRowsPerBlock  = kWavesPerBlock * kRowsPerWave; // 32
constexpr int kGatherLds     = kRowsPerBlock * kRowBytes;     // 38400 B
}

// ----- CDNA5 async global<->LDS helpers (ASYNCcnt-tracked DMA path) --------

__device__ __forceinline__ void async_load_b128(unsigned lds_addr, const void* gaddr) {
  asm volatile("global_load_async_to_lds_b128 %0, %1, off"
               :: "v"(lds_addr), "v"((unsigned long long)gaddr)
               : "memory");
}

// Non-temporal: outputs are streamed once; don't let them evict the
// embedding tables from GL2.
__device__ __forceinline__ void async_store_b128_nt(void* gaddr, unsigned lds_addr) {
  asm volatile("global_store_async_from_lds_b128 %0, %1, off th:TH_STORE_NT"
               :: "v"((unsigned long long)gaddr), "v"(lds_addr)
               : "memory");
}

__device__ __forceinline__ void wait_async0() {
  asm volatile("s_wait_asynccnt 0" ::: "memory");
}

// ----- Kernel 1: out_vec + neg_vec gathers via async LDS DMA ---------------
// One wave per 4 rows: 12 back-to-back async b128 loads (4 rows x 75 chunks
// / 32 lanes), one s_wait_asynccnt 0, 12 async b128 NT stores. VGPRs stay
// free (high occupancy hides HBM latency); each wave-op moves 512B/176B.

__global__ void __launch_bounds__(256)
skipgram_gather(const float* __restrict__ W_out,
                const int*   __restrict__ out_idx,
                const int*   __restrict__ ng_idx,
                float*       __restrict__ out) {
  extern __shared__ char lds_raw[];   // dynamic LDS: kGatherLds bytes
  (void)lds_raw;

  const int lane = threadIdx.x & 31;
  const int wave = threadIdx.x >> 5;
  const int row0 = blockIdx.x * kRowsPerBlock + wave * kRowsPerWave;
  const unsigned lds_wave = (unsigned)(wave * (kRowsPerWave * kRowBytes));
  const bool third = lane < kThirdLanes;

  // Row indices first, then a pure burst of async issue.
  int idx[kRowsPerWave];
#pragma unroll
  for (int r = 0; r < kRowsPerWave; ++r) {
    const int row = row0 + r;
    idx[r] = (row < kOutRows) ? out_idx[row] : ng_idx[row - kOutRows];
  }

  // Phase 1: gather rows from W_out into LDS (async, no VGPR data).
#pragma unroll
  for (int r = 0; r < kRowsPerWave; ++r) {
    const char* src = (const char*)(W_out + (size_t)idx[r] * kEmb);
    const unsigned lbuf = lds_wave + (unsigned)(r * kRowBytes);
    async_load_b128(lbuf + (unsigned)lane * 16u,        src + (size_t)lane * 16);
    async_load_b128(lbuf + (unsigned)(lane + 32) * 16u, src + (size_t)(lane + 32) * 16);
    if (third)
      async_load_b128(lbuf + (unsigned)(lane + 64) * 16u, src + (size_t)(lane + 64) * 16);
  }
  wait_async0();   // LDS now holds all 4 rows

  // Phase 2: stream LDS out to the packed destination (async, NT).
#pragma unroll
  for (int r = 0; r < kRowsPerWave; ++r) {
    const int row = row0 + r;
    char* dst = (char*)((row < kOutRows)
                            ? (out + kOutBase + (size_t)row * kEmb)
                            : (out + kNegBase + (size_t)(row - kOutRows) * kEmb));
    const unsigned lbuf = lds_wave + (unsigned)(r * kRowBytes);
    async_store_b128_nt(dst + (size_t)lane * 16,        lbuf + (unsigned)lane * 16u);
    async_store_b128_nt(dst + (size_t)(lane + 32) * 16, lbuf + (unsigned)(lane + 32) * 16u);
    if (third)
      async_store_b128_nt(dst + (size_t)(lane + 64) * 16, lbuf + (unsigned)(lane + 64) * 16u);
  }
  wait_async0();   // stores have consumed LDS (S_ENDPGM also waits idle)
}

// ----- Kernel 2: segment-summed input embeddings ---------------------------

__device__ __forceinline__ int lower_bound32(const int* __restrict__ a,
                                             int n, int key) {
  int lo = 0, len = n;
  while (len > 0) {
    const int half = len >> 1;
    const int mid  = lo + half;
    if (a[mid] < key) { lo = mid + 1; len -= half + 1; }
    else              { len = half; }
  }
  return lo;
}

__global__ void __launch_bounds__(256)
skipgram_segsum(const float* __restrict__ W_in,
                const int*   __restrict__ flat_idx,
                const int*   __restrict__ seg_ids,
                float*       __restrict__ out) {
  const int lane = threadIdx.x & 31;
  const int wave = threadIdx.x >> 5;
  const int seg  = blockIdx.x * kWavesPerBlock + wave;   // exact cover

  // segment_ids is sorted: binary search this segment's token range.
  const int lo = lower_bound32(seg_ids, kTotal, seg);
  const int hi = lower_bound32(seg_ids, kTotal, seg + 1);

  v4f a0 = {0.f, 0.f, 0.f, 0.f};
  v4f a1 = a0, a2 = a0;
  const bool third = lane < kThirdLanes;   // lanes 0..10 own a third chunk

  for (int t = lo; t < hi; ++t) {
    const v4f* src = (const v4f*)(W_in + (size_t)flat_idx[t] * kEmb);
    if (t + 1 < hi && lane < 19) {   // 19 x 64B covers the 1200B next row
      const char* nxt = (const char*)(W_in + (size_t)flat_idx[t + 1] * kEmb);
      __builtin_prefetch(nxt + lane * 64, 0, 0);   // global_prefetch_b8
    }
    a0 += src[lane];
    a1 += src[lane + 32];
    if (third) a2 += src[lane + 64];
  }

  // in_vec occupies d_out offset 0; write-once -> non-temporal b128 stores.
  v4f* dst = (v4f*)(out + (size_t)seg * kEmb);
  __builtin_nontemporal_store(a0, dst + lane);
  __builtin_nontemporal_store(a1, dst + lane + 32);
  if (third) __builtin_nontemporal_store(a2, dst + lane + 64);
}

// ----- Launch --------------------------------------------------------------

extern "C" void kernel_launch(void* const* d_in, const int* in_sizes, int n_in,
                              void* d_out, int out_size, void* d_ws, size_t ws_size,
                              hipStream_t stream) {
  (void)in_sizes; (void)n_in; (void)out_size; (void)d_ws; (void)ws_size;

  const float* W_in     = (const float*)d_in[0];
  const float* W_out    = (const float*)d_in[1];
  const int*   flat_idx = (const int*)d_in[2];
  const int*   seg_ids  = (const int*)d_in[3];
  const int*   out_idx  = (const int*)d_in[4];
  const int*   ng_idx   = (const int*)d_in[5];
  float*       out      = (float*)d_out;

  // in_vec: 2048 blocks x 8 waves = 16384 segments
  skipgram_segsum<<<kBatch / kWavesPerBlock, 256, 0, stream>>>(
      W_in, flat_idx, seg_ids, out);

  // out_vec + neg_vec: 3072 blocks x 32 rows = 98304 rows
  skipgram_gather<<<kGatherRows / kRowsPerBlock, 256, kGatherLds, stream>>>(
      W_out, out_idx, ng_idx, out);
}